// RoutingLoss_22058952032712
// MI455X (gfx1250) — compile-verified
//
#include <hip/hip_runtime.h>
#include <stdint.h>

#define EPSILON_F 0.02f
#define TILE_ROWS 256
#define TILE_FLOATS (TILE_ROWS * 5) /* 1280 f32 = 5120 B per decision tile */

typedef unsigned int tdm_u32x4 __attribute__((ext_vector_type(4)));
typedef int          tdm_i32x8 __attribute__((ext_vector_type(8)));
typedef int          tdm_i32x4 __attribute__((ext_vector_type(4)));

// Issue one TDM DMA: copy `nfloats` contiguous f32 from global -> LDS.
// D# layout per CDNA5 ISA §8.3/8.4 (1-D tile expressed as dim0 x 1).
__device__ __forceinline__ void tdm_load_tile(const void* gsrc, uint32_t lds_addr,
                                              unsigned nfloats) {
    const uint64_t ga = (uint64_t)gsrc;

    tdm_u32x4 g0;
    g0[0] = 1u;                                   // count=1, user descriptor
    g0[1] = lds_addr;                             // lds_addr [63:32]
    g0[2] = (uint32_t)ga;                         // global_addr [95:64]
    g0[3] = (uint32_t)(ga >> 32) | 0x80000000u;   // global_addr[56:32] | type=2 [127:126]

    tdm_i32x8 g1;
    g1[0] = 0x00020000;                           // workgroup_mask=0, data_size=2 (4B)
    g1[1] = (int)((nfloats & 0xFFFFu) << 16);     // tensor_dim0[15:0] at [63:48]
    g1[2] = (int)((nfloats >> 16) | (1u << 16));  // tensor_dim0[31:16] | tensor_dim1=1
    g1[3] = (int)((nfloats & 0xFFFFu) << 16);     // tensor_dim1 hi=0 | tile_dim0 [127:112]
    g1[4] = 1;                                    // tile_dim1=1, tile_dim2=0
    g1[5] = (int)nfloats;                         // tensor_dim0_stride[31:0]
    g1[6] = 0;                                    // stride hi | dim1_stride lo
    g1[7] = 0;

    tdm_i32x4 g2 = {0, 0, 0, 0};                  // <=2D tensor: groups 2/3 zero
    tdm_i32x4 g3 = {0, 0, 0, 0};
    tdm_i32x8 g4 = {0, 0, 0, 0, 0, 0, 0, 0};      // extra group (clang-23 6-arg form)

    __builtin_amdgcn_tensor_load_to_lds(g0, g1, g2, g3, g4, 0 /*cpol*/);
}

// Per-row loss terms: CE contribution (logsumexp - d[jusm]) and squared error.
__device__ __forceinline__ void row_terms(float d0, float d1, float d2, float d3, float d4,
                                          float4 t, float4 c,
                                          float& ce_acc, float& sq_acc) {
    int j = 0;
    if (t.x >= EPSILON_F) j = 1;
    if (t.y >= EPSILON_F) j = 2;
    if (t.z >= EPSILON_F) j = 3;
    if (t.w >= EPSILON_F) j = 4;

    float ex = c.x - t.x, ey = c.y - t.y, ez = c.z - t.z, ew = c.w - t.w;
    sq_acc += ex * ex + ey * ey + ez * ez + ew * ew;

    float m = fmaxf(fmaxf(fmaxf(d0, d1), fmaxf(d2, d3)), d4);
    float s = expf(d0 - m) + expf(d1 - m) + expf(d2 - m) + expf(d3 - m) + expf(d4 - m);
    float lse = m + logf(s);

    float sel = d0;
    if (j == 1) sel = d1;
    if (j == 2) sel = d2;
    if (j == 3) sel = d3;
    if (j == 4) sel = d4;
    ce_acc += lse - sel;
}

__global__ __launch_bounds__(256)
void routing_partial_kernel(const float* __restrict__ dec,
                            const float4* __restrict__ cost4,
                            const float4* __restrict__ tgt4,
                            float* __restrict__ partials,
                            int rows) {
    __shared__ float sdec[2][TILE_FLOATS];  // double-buffered decision tiles
    __shared__ float rce[8], rsq[8];

    const int tid = threadIdx.x;
    float ce = 0.0f, sq = 0.0f;

    const int tiles = rows / TILE_ROWS;
    const int b = (int)blockIdx.x;
    const int stride = (int)gridDim.x;
    const int myTiles = (tiles > b) ? ((tiles - 1 - b) / stride + 1) : 0;

    const uint32_t lds[2] = { (uint32_t)(size_t)(&sdec[0][0]),
                              (uint32_t)(size_t)(&sdec[1][0]) };

    // Prefetch first tile via TDM (wave 0 only -> single DMA per block).
    if (tid < 32 && myTiles > 0) {
        tdm_load_tile((const char*)dec + (size_t)b * TILE_ROWS * 20u, lds[0], TILE_FLOATS);
    }

    for (int m = 0; m < myTiles; ++m) {
        const int tile = b + m * stride;
        const size_t rowBase = (size_t)tile * TILE_ROWS;
        const bool hasNext = (m + 1) < myTiles;

        if (tid < 32) {
            if (hasNext) {
                // Overlap: DMA tile m+1 into the other buffer while we compute tile m.
                const size_t nextBase = rowBase + (size_t)stride * TILE_ROWS;
                tdm_load_tile((const char*)dec + nextBase * 20u, lds[(m + 1) & 1],
                              TILE_FLOATS);
                // TDM ops from one wave complete in order: <=1 outstanding means
                // tile m has fully landed in LDS.
                __builtin_amdgcn_s_wait_tensorcnt(1);
            } else {
                __builtin_amdgcn_s_wait_tensorcnt(0);
            }
        }
        __syncthreads();  // publish tile m to all waves

        {
            const float* buf = &sdec[m & 1][0];
            const int r = tid;
            const int i = (int)rowBase + r;
            float4 t = tgt4[i];   // global_load_b128, coalesced
            float4 c = cost4[i];  // global_load_b128, coalesced
            // stride-5 dword reads: gcd(5,64)=1 -> bank-conflict free
            float d0 = buf[r * 5 + 0];
            float d1 = buf[r * 5 + 1];
            float d2 = buf[r * 5 + 2];
            float d3 = buf[r * 5 + 3];
            float d4 = buf[r * 5 + 4];
            row_terms(d0, d1, d2, d3, d4, t, c, ce, sq);
        }
        __syncthreads();  // reads of buf[m&1] done before it is reused as dest
    }

    // Tail rows (rows % TILE_ROWS) via direct loads; empty for N = 4M.
    const int tailStart = tiles * TILE_ROWS;
    if (tailStart < rows && blockIdx.x == 0) {
        for (int i = tailStart + tid; i < rows; i += (int)blockDim.x) {
            const size_t base = (size_t)i * 5u;
            float d0 = dec[base + 0], d1 = dec[base + 1], d2 = dec[base + 2],
                  d3 = dec[base + 3], d4 = dec[base + 4];
            row_terms(d0, d1, d2, d3, d4, tgt4[i], cost4[i], ce, sq);
        }
    }

    // wave32 shuffle reduction, then cross-wave via LDS (8 waves / block)
    const int lane = tid & 31, wv = tid >> 5;
    for (int off = 16; off > 0; off >>= 1) {
        ce += __shfl_down(ce, off, 32);
        sq += __shfl_down(sq, off, 32);
    }
    if (lane == 0) { rce[wv] = ce; rsq[wv] = sq; }
    __syncthreads();
    if (tid == 0) {
        float tce = 0.0f, tsq = 0.0f;
        for (int w = 0; w < (int)(blockDim.x >> 5); ++w) { tce += rce[w]; tsq += rsq[w]; }
        partials[2 * blockIdx.x + 0] = tce;
        partials[2 * blockIdx.x + 1] = tsq;
    }
}

__global__ __launch_bounds__(256)
void routing_finalize_kernel(const float* __restrict__ partials, int nb,
                             float* __restrict__ out, int rows) {
    __shared__ double sce[256], ssq[256];
    double ce = 0.0, sq = 0.0;
    for (int i = threadIdx.x; i < nb; i += (int)blockDim.x) {
        ce += (double)partials[2 * i + 0];
        sq += (double)partials[2 * i + 1];
    }
    sce[threadIdx.x] = ce;
    ssq[threadIdx.x] = sq;
    __syncthreads();
    for (int off = 128; off > 0; off >>= 1) {
        if ((int)threadIdx.x < off) {
            sce[threadIdx.x] += sce[threadIdx.x + off];
            ssq[threadIdx.x] += ssq[threadIdx.x + off];
        }
        __syncthreads();
    }
    if (threadIdx.x == 0) {
        const double n = (double)rows;
        // loss = CE_mean * (1-GAMMA) + MSE_mean * GAMMA, GAMMA = 0.5
        const double loss = (sce[0] / n) * 0.5 + (ssq[0] / (n * 4.0)) * 0.5;
        out[0] = (float)loss;
    }
}

extern "C" void kernel_launch(void* const* d_in, const int* in_sizes, int n_in,
                              void* d_out, int out_size, void* d_ws, size_t ws_size,
                              hipStream_t stream) {
    const float*  dec   = (const float*)d_in[0];   // [N,5] f32
    const float4* cost4 = (const float4*)d_in[1];  // [N,4] f32, 16B rows
    const float4* tgt4  = (const float4*)d_in[2];  // [N,4] f32, 16B rows
    const int rows = in_sizes[0] / 5;

    int NB = 4096;  // 4096 blocks x 8 waves -> deep VMEM/TDM pipelines
    while ((size_t)NB * 2u * sizeof(float) > ws_size && NB > 1) NB >>= 1;

    float* partials = (float*)d_ws;
    routing_partial_kernel<<<dim3(NB), dim3(256), 0, stream>>>(dec, cost4, tgt4, partials, rows);
    routing_finalize_kernel<<<dim3(1), dim3(256), 0, stream>>>(partials, NB, (float*)d_out, rows);
}